// ProtoNet_38826504356173
// MI455X (gfx1250) — compile-verified
//
#include <hip/hip_runtime.h>

typedef __attribute__((ext_vector_type(2))) float v2f;
typedef __attribute__((ext_vector_type(4))) float v4f;
typedef __attribute__((ext_vector_type(8))) float v8f;

#define B_      2048
#define P_      50
#define K_      28672          // 512*7*8
#define NPAD    64
#define KSPLIT  8
#define KPER    (K_ / KSPLIT)  // 3584
#define KC      256            // K chunk staged in LDS
#define LDSS    260            // padded LDS row stride (floats), bank-conflict free

// workspace layout in floats:
//   xp slabs: [KSPLIT][B_][NPAD]   -> 8*2048*64 = 1,048,576 floats
//   x2 slabs: [KSPLIT][B_]         -> 16,384 floats
//   p2      : [NPAD]               -> 64 floats
#define WS_XP   0
#define WS_X2   (KSPLIT * B_ * NPAD)
#define WS_P2   (WS_X2 + KSPLIT * B_)

// -------- prototype squared norms --------------------------------------
__global__ __launch_bounds__(256) void p2_kernel(const float* __restrict__ proto,
                                                 float* __restrict__ ws) {
    const int p = blockIdx.x;          // 0..63 (>=P_ writes 0)
    float s = 0.f;
    if (p < P_) {
        const float* row = proto + (size_t)p * K_;
        for (int k = threadIdx.x; k < K_; k += 256) {
            float v = row[k];
            s += v * v;
        }
    }
    __shared__ float red[256];
    red[threadIdx.x] = s;
    __syncthreads();
    for (int w = 128; w > 0; w >>= 1) {
        if (threadIdx.x < w) red[threadIdx.x] += red[threadIdx.x + w];
        __syncthreads();
    }
    if (threadIdx.x == 0) ws[WS_P2 + p] = red[0];
}

// -------- main GEMM: xp[b,p] = sum_k x[b,k]*proto[p,k] (f32 WMMA) ------
__global__ __launch_bounds__(128) void gemm_kernel(const float* __restrict__ x,
                                                   const float* __restrict__ proto,
                                                   float* __restrict__ ws) {
    __shared__ float ldsX[16 * LDSS];

    const int mtile = blockIdx.x;        // 0..127  (16 rows of x each)
    const int ks    = blockIdx.y;        // 0..KSPLIT-1
    const int tid   = threadIdx.x;       // 0..127
    const int lane  = tid & 31;
    const int wave  = tid >> 5;          // 0..3 -> N-tile (16 protos each)

    const int m    = lane & 15;          // A row within tile / B column
    const int koff = (lane >> 4) << 1;   // lane-half selects K {0,1} vs {2,3}
    const int k0   = ks * KPER;

    int p = wave * 16 + m;
    p = p < P_ ? p : (P_ - 1);           // clamp padding columns (results unused)
    const float* bbase = proto + (size_t)p * K_ + k0 + koff;
    const float* xbase = x + (size_t)(mtile * 16) * K_ + k0;

    // this thread's 8 float4 staging slots (fixed across chunks)
    const int row0 = tid >> 6;           // base row contribution (q = i*128 + tid)
    const int c4   = (tid & 63) * 4;     // float offset within row

    v8f acc = {};
    float s2 = 0.f;                      // per-lane partial of sum(x^2)

    // prologue: prefetch x chunk 0 into registers (non-temporal: x is stream-once,
    // keep the L2 for the prototype matrix which every workgroup re-reads)
    v4f pre[8];
    #pragma unroll
    for (int i = 0; i < 8; ++i) {
        int row = i * 2 + row0;
        pre[i] = __builtin_nontemporal_load(
            (const v4f*)(xbase + (size_t)row * K_ + c4));
    }

    // prologue: preload first B group (8 k-steps = 32 floats of this lane's proto)
    v2f bb[8];
    #pragma unroll
    for (int j = 0; j < 8; ++j)
        bb[j] = *(const v2f*)(bbase + j * 4);

    for (int kb = 0; kb < KPER; kb += KC) {
        __syncthreads();                 // previous chunk's readers done with LDS
        #pragma unroll
        for (int i = 0; i < 8; ++i) {
            int row = i * 2 + row0;
            *(v4f*)(&ldsX[row * LDSS + c4]) = pre[i];
        }
        __syncthreads();

        // issue next x chunk's global loads now; latency overlaps the WMMA phase
        if (kb + KC < KPER) {
            #pragma unroll
            for (int i = 0; i < 8; ++i) {
                int row = i * 2 + row0;
                pre[i] = __builtin_nontemporal_load(
                    (const v4f*)(xbase + (size_t)row * K_ + kb + KC + c4));
            }
        }

        // compute: 8 groups of 8 k-steps; B pipelined one group ahead so the
        // load clause issued here is only waited on at the NEXT group's WMMAs
        #pragma unroll 1
        for (int g = 0; g < KC / 32; ++g) {
            int knext = kb + (g + 1) * 32;                    // contiguous across chunks
            knext = knext <= (KPER - 32) ? knext : (KPER - 32); // in-bounds tail clamp
            v2f bn[8];
            #pragma unroll
            for (int j = 0; j < 8; ++j)
                bn[j] = *(const v2f*)(bbase + knext + j * 4);  // global_load_b64 clause

            #pragma unroll
            for (int j = 0; j < 8; ++j) {
                v2f a = *(const v2f*)(&ldsX[m * LDSS + koff + g * 32 + j * 4]);
                s2 += a.x * a.x + a.y * a.y;                  // co-executes with XDL
                acc = __builtin_amdgcn_wmma_f32_16x16x4_f32(
                    false, a, false, bb[j], (short)0, acc, false, false);
            }
            #pragma unroll
            for (int j = 0; j < 8; ++j) bb[j] = bn[j];
        }
    }

    // store xp partial tile: C layout lane L: N=L%16, M=r+8*(L/16)
    {
        const int mb = (lane >> 4) * 8;
        float* dst = ws + WS_XP + ((size_t)ks * B_ + mtile * 16) * NPAD + wave * 16 + m;
        #pragma unroll
        for (int r = 0; r < 8; ++r) dst[(size_t)(mb + r) * NPAD] = acc[r];
    }

    // x2 partial: lane halves together cover all K of this split
    s2 += __shfl_xor(s2, 16, 32);
    if (wave == 0 && lane < 16)
        ws[WS_X2 + ks * B_ + mtile * 16 + lane] = s2;
}

// -------- finalize: relu(x2 + p2 - 2*xp) -------------------------------
__global__ __launch_bounds__(256) void fin_kernel(const float* __restrict__ ws,
                                                  float* __restrict__ out) {
    int t = blockIdx.x * 256 + threadIdx.x;
    if (t >= B_ * P_) return;
    int b = t / P_;
    int p = t - b * P_;
    float xp = 0.f, x2 = 0.f;
    #pragma unroll
    for (int ks = 0; ks < KSPLIT; ++ks) {
        xp += ws[WS_XP + ((size_t)ks * B_ + b) * NPAD + p];
        x2 += ws[WS_X2 + ks * B_ + b];
    }
    float d = x2 + ws[WS_P2 + p] - 2.f * xp;
    out[t] = d > 0.f ? d : 0.f;
}

extern "C" void kernel_launch(void* const* d_in, const int* in_sizes, int n_in,
                              void* d_out, int out_size, void* d_ws, size_t ws_size,
                              hipStream_t stream) {
    (void)in_sizes; (void)n_in; (void)out_size; (void)ws_size;
    const float* x     = (const float*)d_in[0];
    const float* proto = (const float*)d_in[1];
    float* ws  = (float*)d_ws;
    float* out = (float*)d_out;

    p2_kernel<<<NPAD, 256, 0, stream>>>(proto, ws);

    dim3 grid(B_ / 16, KSPLIT);
    gemm_kernel<<<grid, 128, 0, stream>>>(x, proto, ws);

    fin_kernel<<<(B_ * P_ + 255) / 256, 256, 0, stream>>>(ws, out);
}